// KeypointMatching_77214922047590
// MI455X (gfx1250) — compile-verified
//
#include <hip/hip_runtime.h>
#include <hip/hip_bf16.h>

// ---------------------------------------------------------------------------
// KeypointMatching on MI455X (gfx1250, wave32)
//   N=50000, K=32 neighbors, C=128 channels, top-16 column-global mask.
// Memory-bound: 2 passes over knn_feat (819MB each) dominate (~73us floor).
// fp32 WMMA (V_WMMA_F32_16X16X4_F32) used for S = feat @ [Wq^T Wk | W_sym].
// ---------------------------------------------------------------------------

typedef float v2f __attribute__((ext_vector_type(2)));
typedef float v8f __attribute__((ext_vector_type(8)));

#define CC   128
#define KK   32
#define NEGB 1e12f

// ---------------------------------------------------------------------------
// Kernel 0: Bmat[c][0:128]   = (Wq^T Wk)[c][j]  = sum_d Wq[d][c]*Wk[d][j]
//           Bmat[c][128:256] = W_sym[c][j]      = triu(W)+triu(W)^T
//           zero the global 32-bit column-survival mask
// grid: 128 blocks x 256 threads
// ---------------------------------------------------------------------------
__global__ void __launch_bounds__(256) km_prep(
    const float* __restrict__ Wq, const float* __restrict__ Wk,
    const float* __restrict__ W, float* __restrict__ Bmat,
    unsigned int* __restrict__ colMask) {
  const int c = blockIdx.x;    // 0..127
  const int j = threadIdx.x;   // 0..255
  float v;
  if (j < CC) {
    float acc = 0.f;
    #pragma unroll 8
    for (int d = 0; d < CC; ++d) acc += Wq[d * CC + c] * Wk[d * CC + j];
    v = acc;
  } else {
    const int jj = j - CC;
    const float a = (c <= jj) ? W[c * CC + jj] : 0.f;
    const float b = (jj <= c) ? W[jj * CC + c] : 0.f;
    v = a + b;
  }
  Bmat[c * 256 + j] = v;
  if (c == 0 && j == 0) *colMask = 0u;
}

// ---------------------------------------------------------------------------
// Kernel 1: per 16-row tile:
//   Phase A: S[16][256] = feat_tile[16][128] @ Bmat[128][256] via f32 WMMA.
//            wave w owns column tiles {2w, 2w+1}; 32 k-steps of K=4.
//   Phase B: wave w scores rows {2w, 2w+1}: lane = neighbor k.
//            scores -> ws, match_logits -> d_out, top-16 bits -> colMask.
// ---------------------------------------------------------------------------
__global__ void __launch_bounds__(256) km_score(
    const float* __restrict__ feat, const float* __restrict__ knn_feat,
    const unsigned char* __restrict__ knn_mask,
    const float* __restrict__ Bmat,
    float* __restrict__ scores, float* __restrict__ match_logits,
    unsigned int* __restrict__ colMask, int N) {
  __shared__ float S[16 * 256];
  __shared__ unsigned int blkMask;

  const int tid  = threadIdx.x;
  const int wave = tid >> 5;
  const int lane = tid & 31;
  if (tid == 0) blkMask = 0u;

  const int row0    = blockIdx.x * 16;
  const int halfsel = lane >> 4;        // which half-wave (K split)
  const int lane16  = lane & 15;

  // ---- Phase A: WMMA f32 16x16x4, accumulate over 128 in chunks of 4 ----
  int arow = row0 + lane16;
  if (arow >= N) arow = N - 1;          // clamp (stores guarded in phase B)
  const float* afp = feat + (size_t)arow * CC;

  v8f acc0 = {};
  v8f acc1 = {};
  const int ct0 = wave * 2;
  const int ct1 = wave * 2 + 1;

  #pragma unroll 4
  for (int kk = 0; kk < 32; ++kk) {
    const int kbase = kk * 4 + halfsel * 2;   // A: VGPR0=K{0,2}, VGPR1=K{1,3}
    v2f a;
    a.x = afp[kbase + 0];
    a.y = afp[kbase + 1];
    v2f b0, b1;                               // B: same K-split across halves
    b0.x = Bmat[(kbase + 0) * 256 + ct0 * 16 + lane16];
    b0.y = Bmat[(kbase + 1) * 256 + ct0 * 16 + lane16];
    b1.x = Bmat[(kbase + 0) * 256 + ct1 * 16 + lane16];
    b1.y = Bmat[(kbase + 1) * 256 + ct1 * 16 + lane16];
    acc0 = __builtin_amdgcn_wmma_f32_16x16x4_f32(
        false, a, false, b0, (short)0, acc0, false, false);
    acc1 = __builtin_amdgcn_wmma_f32_16x16x4_f32(
        false, a, false, b1, (short)0, acc1, false, false);
  }
  // D layout: VGPR r -> row (r + 8*halfsel), col lane16
  #pragma unroll
  for (int r = 0; r < 8; ++r) {
    S[(r + 8 * halfsel) * 256 + ct0 * 16 + lane16] = acc0[r];
    S[(r + 8 * halfsel) * 256 + ct1 * 16 + lane16] = acc1[r];
  }
  __syncthreads();

  // ---- Phase B: scoring; lane = neighbor index ----
  for (int rr = 0; rr < 2; ++rr) {
    const int rloc = wave * 2 + rr;
    const int row  = row0 + rloc;
    if (row >= N) continue;               // uniform per wave
    const float* kf = knn_feat + ((size_t)row * KK + lane) * CC;
    const float* sv = &S[rloc * 256];     // s_vec = sv[0:128], fW = sv[128:256]
    float sacc = 0.f, macc = 0.f;
    #pragma unroll 4
    for (int c = 0; c < CC; c += 4) {
      __builtin_prefetch(kf + c + 32, 0, 0);
      const float4 v = *(const float4*)(kf + c);
      sacc += v.x * sv[c + 0] + v.y * sv[c + 1] +
              v.z * sv[c + 2] + v.w * sv[c + 3];
      macc += v.x * sv[CC + c + 0] + v.y * sv[CC + c + 1] +
              v.z * sv[CC + c + 2] + v.w * sv[CC + c + 3];
    }
    const bool valid = knn_mask[(size_t)row * KK + lane] != 0;
    const float score = sacc - (valid ? 0.f : NEGB);
    scores[(size_t)row * KK + lane]        = score;
    match_logits[(size_t)row * KK + lane]  = macc;

    // exact top-16 membership (stable tie-break by index, like lax.top_k)
    int cnt = 0;
    #pragma unroll 8
    for (int j = 0; j < KK; ++j) {
      const float sj = __shfl(score, j, 32);
      cnt += (sj > score) || (sj == score && j < lane);
    }
    unsigned int bit = (cnt < 16) ? (1u << lane) : 0u;
    #pragma unroll
    for (int off = 16; off >= 1; off >>= 1) bit |= __shfl_xor(bit, off, 32);
    if (lane == 0) atomicOr(&blkMask, bit);
  }
  __syncthreads();
  if (tid == 0) atomicOr(colMask, blkMask);
}

// ---------------------------------------------------------------------------
// Kernel 2: wave per row. Apply column-global mask, softmax, weighted sums.
//   logit = sum_k attn_k * match_logits_k   (exact identity with fW.corres)
// ---------------------------------------------------------------------------
__global__ void __launch_bounds__(256) km_final(
    const float* __restrict__ feat, const float* __restrict__ knn_xyz,
    const float* __restrict__ knn_feat,
    const float* __restrict__ scores, const float* __restrict__ match_logits,
    const unsigned int* __restrict__ colMask,
    float* __restrict__ out_xyz, float* __restrict__ out_af, int N) {
  const int wave = threadIdx.x >> 5;
  const int lane = threadIdx.x & 31;
  const int n = blockIdx.x * 8 + wave;
  if (n >= N) return;                      // uniform per wave

  const unsigned int cm = *colMask;
  const float score = scores[(size_t)n * KK + lane];
  const float ml    = match_logits[(size_t)n * KK + lane];
  const bool  alive = (cm >> lane) & 1u;

  // softmax over 32 with -inf on dead columns
  float m = alive ? score : -3.402823466e38f;
  #pragma unroll
  for (int off = 16; off >= 1; off >>= 1) m = fmaxf(m, __shfl_xor(m, off, 32));
  const float e = alive ? __expf(score - m) : 0.f;
  float denom = e;
  #pragma unroll
  for (int off = 16; off >= 1; off >>= 1) denom += __shfl_xor(denom, off, 32);
  const float attn = e / denom;

  // corres_xyz + logit (cross-lane reductions)
  const float* xp = knn_xyz + ((size_t)n * KK + lane) * 3;
  float x = attn * xp[0];
  float y = attn * xp[1];
  float z = attn * xp[2];
  float lg = attn * ml;
  #pragma unroll
  for (int off = 16; off >= 1; off >>= 1) {
    x  += __shfl_xor(x, off, 32);
    y  += __shfl_xor(y, off, 32);
    z  += __shfl_xor(z, off, 32);
    lg += __shfl_xor(lg, off, 32);
  }
  if (lane == 0) {
    out_xyz[(size_t)n * 3 + 0] = x;
    out_xyz[(size_t)n * 3 + 1] = y;
    out_xyz[(size_t)n * 3 + 2] = z;
  }

  // corres_feat: lane owns channels [4*lane, 4*lane+4)
  float a0 = 0.f, a1 = 0.f, a2 = 0.f, a3 = 0.f;
  const float* kf = knn_feat + (size_t)n * KK * CC;
  #pragma unroll 4
  for (int k = 0; k < KK; ++k) {
    __builtin_prefetch(kf + (k + 4) * CC + lane * 4, 0, 0);
    const float ak = __shfl(attn, k, 32);
    const float4 v = *(const float4*)(kf + k * CC + lane * 4);
    a0 += ak * v.x; a1 += ak * v.y; a2 += ak * v.z; a3 += ak * v.w;
  }

  float* afrow = out_af + (size_t)n * 257;       // rows of 257 -> scalar stores
  const float4 fv = *(const float4*)(feat + (size_t)n * CC + lane * 4);
  afrow[lane * 4 + 0] = fv.x;
  afrow[lane * 4 + 1] = fv.y;
  afrow[lane * 4 + 2] = fv.z;
  afrow[lane * 4 + 3] = fv.w;
  afrow[CC + lane * 4 + 0] = a0;
  afrow[CC + lane * 4 + 1] = a1;
  afrow[CC + lane * 4 + 2] = a2;
  afrow[CC + lane * 4 + 3] = a3;
  if (lane == 0) afrow[256] = lg;
}

// ---------------------------------------------------------------------------
// Launch: prep -> score (WMMA) -> final, all on `stream`.
// ws layout: Bmat (128*256 f32) | scores (N*32 f32) | colMask (u32)
// d_out layout: corres_xyz[N*3] | attentive_feats[N*257] | match_logits[N*32]
// ---------------------------------------------------------------------------
extern "C" void kernel_launch(void* const* d_in, const int* in_sizes, int n_in,
                              void* d_out, int out_size, void* d_ws, size_t ws_size,
                              hipStream_t stream) {
  const float*         feat     = (const float*)d_in[0];
  const float*         knn_xyz  = (const float*)d_in[1];
  const float*         knn_feat = (const float*)d_in[2];
  const unsigned char* knn_mask = (const unsigned char*)d_in[3];
  const float*         Wq       = (const float*)d_in[4];
  const float*         Wk       = (const float*)d_in[5];
  const float*         W        = (const float*)d_in[6];

  const int N = in_sizes[0] / CC;

  float* out     = (float*)d_out;
  float* out_xyz = out;
  float* out_af  = out + (size_t)N * 3;
  float* out_ml  = out + (size_t)N * 3 + (size_t)N * 257;

  char* ws = (char*)d_ws;
  float*        Bmat    = (float*)ws;                                   // 128KB
  float*        scoresW = (float*)(ws + 128 * 256 * sizeof(float));     // N*32 f32
  unsigned int* colMask = (unsigned int*)(ws + 128 * 256 * sizeof(float) +
                                          (size_t)N * KK * sizeof(float));

  km_prep<<<CC, 256, 0, stream>>>(Wq, Wk, W, Bmat, colMask);
  km_score<<<(N + 15) / 16, 256, 0, stream>>>(feat, knn_feat, knn_mask, Bmat,
                                              scoresW, out_ml, colMask, N);
  km_final<<<(N + 7) / 8, 256, 0, stream>>>(feat, knn_xyz, knn_feat, scoresW,
                                            out_ml, colMask, out_xyz, out_af, N);
}